// TexLossOptTrans_23648089931988
// MI455X (gfx1250) — compile-verified
//
#include <hip/hip_runtime.h>
#include <hip/hip_bf16.h>

typedef __attribute__((ext_vector_type(2))) float v2f;
typedef __attribute__((ext_vector_type(8))) float v8f;

static constexpr int   N        = 512;       // a*b
static constexpr int   CD       = 65536;     // c*d
static constexpr float EPSV     = 1e-6f;
static constexpr int   NS_ITERS = 15;
static constexpr int   KS       = 4;         // split-K slices for the Gram

__device__ __forceinline__ v8f wmma_f32(v2f a, v2f b, v8f c) {
  // V_WMMA_F32_16X16X4_F32 : D = A(16x4) * B(4x16) + C(16x16), f32 path
  return __builtin_amdgcn_wmma_f32_16x16x4_f32(
      /*neg_a=*/false, a, /*neg_b=*/false, b,
      /*c_mod=*/(short)0, c, /*reuse_a=*/false, /*reuse_b=*/false);
}

// ---------------------------------------------------------------- row means
// grid (512, 2): y selects matrix.
__global__ void __launch_bounds__(256)
row_mean_kernel(const float* __restrict__ X0, const float* __restrict__ X1,
                float* __restrict__ M0, float* __restrict__ M1) {
  const float* X = blockIdx.y ? X1 : X0;
  float*       M = blockIdx.y ? M1 : M0;
  const int row = blockIdx.x;
  const float4* x = (const float4*)(X + (size_t)row * CD);
  float s = 0.f;
  for (int i = threadIdx.x; i < CD / 4; i += 256) {
    float4 v = x[i];
    s += v.x + v.y + v.z + v.w;
  }
  __shared__ float red[256];
  red[threadIdx.x] = s;
  __syncthreads();
  for (int off = 128; off > 0; off >>= 1) {
    if ((int)threadIdx.x < off) red[threadIdx.x] += red[threadIdx.x + off];
    __syncthreads();
  }
  if (threadIdx.x == 0) M[row] = red[0] * (1.f / CD);
}

// ---------------------------------------------- Gram partial: S = X[:,slice] X[:,slice]^T
// grid (8, 8, 2*KS): z = m*KS + slice. WG = 4 waves, 64x64 tile; wave w owns 16 rows.
// Writes raw partial products; fix-up/reduction is a separate deterministic pass.
__global__ void __launch_bounds__(128)
gram_partial_kernel(const float* __restrict__ X0, const float* __restrict__ X1,
                    float* __restrict__ Spart) {
  const int m     = blockIdx.z / KS;
  const int slice = blockIdx.z % KS;
  const float* X = m ? X1 : X0;

  const int wave = threadIdx.x >> 5;
  const int lane = threadIdx.x & 31;
  const int half = lane >> 4;
  const int r    = lane & 15;
  const int rbase = blockIdx.y * 64 + wave * 16;
  const int cbase = blockIdx.x * 64;
  const int kbeg = slice * (CD / KS);
  const int kend = kbeg + (CD / KS);

  v8f acc[4] = {};
  const size_t aoff = (size_t)(rbase + r) * CD + 2 * half;
  size_t boff[4];
#pragma unroll
  for (int t = 0; t < 4; ++t)
    boff[t] = (size_t)(cbase + t * 16 + r) * CD + 2 * half;

  for (int k0 = kbeg; k0 < kend; k0 += 32) {
    // stream ~8KB ahead; speculative prefetch, OOB at the tail is dropped by HW
    __builtin_prefetch(X + aoff + k0 + 2048, 0, 1);
#pragma unroll
    for (int t = 0; t < 4; ++t)
      __builtin_prefetch(X + boff[t] + k0 + 2048, 0, 1);
#pragma unroll
    for (int kk = 0; kk < 32; kk += 4) {
      const int k = k0 + kk;
      v2f a = *(const v2f*)(X + aoff + k);
#pragma unroll
      for (int t = 0; t < 4; ++t) {
        v2f b = *(const v2f*)(X + boff[t] + k);  // B = X^T -> row load of X
        acc[t] = wmma_f32(a, b, acc[t]);
      }
    }
  }

  float* Sp = Spart + (size_t)blockIdx.z * N * N;
#pragma unroll
  for (int t = 0; t < 4; ++t)
#pragma unroll
    for (int v = 0; v < 8; ++v) {
      const int grow = rbase + v + 8 * half;
      const int gcol = cbase + t * 16 + r;
      Sp[(size_t)grow * N + gcol] = acc[t][v];
    }
}

// ---------------------------- reduce split-K partials, apply mean/eps fix-up,
// emit per-tile Frobenius-sumsq and trace partials. grid (64, 2), 256 thr.
__global__ void __launch_bounds__(256)
gram_fixup_kernel(const float* __restrict__ Spart,
                  const float* __restrict__ M0, const float* __restrict__ M1,
                  float* __restrict__ G0, float* __restrict__ G1,
                  float* __restrict__ ps0, float* __restrict__ pt0,
                  float* __restrict__ ps1, float* __restrict__ pt1) {
  const int m    = blockIdx.y;
  const int tile = blockIdx.x;            // 8x8 grid of 64x64 tiles
  const int trow = tile >> 3, tcol = tile & 7;
  const float* Mm = m ? M1 : M0;
  float* G  = m ? G1  : G0;
  float* ps = m ? ps1 : ps0;
  float* pt = m ? pt1 : pt0;

  const float inv = 1.f / CD;
  float ssq = 0.f, trace = 0.f;
  for (int e = threadIdx.x; e < 64 * 64; e += 256) {
    const int row = trow * 64 + (e >> 6);
    const int col = tcol * 64 + (e & 63);
    const size_t idx = (size_t)row * N + col;
    float s = 0.f;
#pragma unroll
    for (int sl = 0; sl < KS; ++sl)
      s += Spart[(size_t)(m * KS + sl) * N * N + idx];
    float val = s * inv - Mm[row] * Mm[col];
    if (row == col) { val += EPSV * inv; trace += val; }
    G[idx] = val;
    ssq += val * val;
  }

  __shared__ float red[256];
  red[threadIdx.x] = ssq;
  __syncthreads();
  for (int off = 128; off > 0; off >>= 1) {
    if ((int)threadIdx.x < off) red[threadIdx.x] += red[threadIdx.x + off];
    __syncthreads();
  }
  if (threadIdx.x == 0) ps[tile] = red[0];
  __syncthreads();
  red[threadIdx.x] = trace;
  __syncthreads();
  for (int off = 128; off > 0; off >>= 1) {
    if ((int)threadIdx.x < off) red[threadIdx.x] += red[threadIdx.x + off];
    __syncthreads();
  }
  if (threadIdx.x == 0) pt[tile] = red[0];
}

// -------------------------------------------------- stats: out[0]=||G||_F, out[1]=tr(G)
__global__ void __launch_bounds__(64)
finalize_stats_kernel(const float* __restrict__ ps, const float* __restrict__ pt,
                      float* __restrict__ out) {
  __shared__ float red[64];
  const int t = threadIdx.x;
  red[t] = ps[t];
  __syncthreads();
  for (int off = 32; off > 0; off >>= 1) {
    if (t < off) red[t] += red[t + off];
    __syncthreads();
  }
  if (t == 0) out[0] = sqrtf(red[0]);
  __syncthreads();
  red[t] = pt[t];
  __syncthreads();
  for (int off = 32; off > 0; off >>= 1) {
    if (t < off) red[t] += red[t + off];
    __syncthreads();
  }
  if (t == 0) out[1] = red[0];
}

// -------------------------------------------------- Y = G/||G||_F ; Z = I
__global__ void __launch_bounds__(256)
init_yz_kernel(const float* __restrict__ G, const float* __restrict__ stats,
               float* __restrict__ Y, float* __restrict__ Z) {
  const int idx = blockIdx.x * 256 + threadIdx.x;
  const float invn = 1.f / stats[0];
  Y[idx] = G[idx] * invn;
  Z[idx] = ((idx >> 9) == (idx & (N - 1))) ? 1.f : 0.f;
}

// -------------------------------------------------- 512^3 WMMA GEMM
// mode 0: D = 1.5*I - 0.5*(A@B)   (the Newton-Schulz T update)
// mode 1: D = A@B
// One wave per block; 32x32 register tile (2x2 WMMA tiles).
__global__ void __launch_bounds__(32)
gemm512_kernel(const float* __restrict__ A, const float* __restrict__ B,
               float* __restrict__ D, int mode) {
  const int lane = threadIdx.x & 31;
  const int half = lane >> 4;
  const int r    = lane & 15;
  const int rb = blockIdx.y * 32;
  const int cb = blockIdx.x * 32;

  v8f acc[2][2] = {};
#pragma unroll 2
  for (int k = 0; k < N; k += 4) {
    v2f a[2], b[2];
#pragma unroll
    for (int i = 0; i < 2; ++i)
      a[i] = *(const v2f*)(A + (size_t)(rb + i * 16 + r) * N + k + 2 * half);
#pragma unroll
    for (int j = 0; j < 2; ++j) {
      v2f t;
      t.x = B[(size_t)(k + 2 * half)     * N + cb + j * 16 + r];
      t.y = B[(size_t)(k + 2 * half + 1) * N + cb + j * 16 + r];
      b[j] = t;
    }
#pragma unroll
    for (int i = 0; i < 2; ++i)
#pragma unroll
      for (int j = 0; j < 2; ++j)
        acc[i][j] = wmma_f32(a[i], b[j], acc[i][j]);
  }

#pragma unroll
  for (int i = 0; i < 2; ++i)
#pragma unroll
    for (int j = 0; j < 2; ++j)
#pragma unroll
      for (int v = 0; v < 8; ++v) {
        const int row = rb + i * 16 + v + 8 * half;
        const int col = cb + j * 16 + r;
        float val = acc[i][j][v];
        if (mode == 0) val = -0.5f * val + ((row == col) ? 1.5f : 0.f);
        D[(size_t)row * N + col] = val;
      }
}

// -------------------------------------------------- sum((M-Mt)^2)
__global__ void __launch_bounds__(512)
mse_kernel(const float* __restrict__ M, const float* __restrict__ Mt,
           float* __restrict__ s_out) {
  __shared__ float red[512];
  const int t = threadIdx.x;
  const float d = M[t] - Mt[t];
  red[t] = d * d;
  __syncthreads();
  for (int off = 256; off > 0; off >>= 1) {
    if (t < off) red[t] += red[t + off];
    __syncthreads();
  }
  if (t == 0) *s_out = red[0];
}

// -------------------------------------------------- sum_{i,k} Yg[i,k]*Ygt[k,i]
__global__ void __launch_bounds__(1024)
cross_trace_kernel(const float* __restrict__ Yg, const float* __restrict__ Ygt,
                   float* __restrict__ s_out) {
  __shared__ float red[1024];
  const int t = threadIdx.x;
  float s = 0.f;
  for (int idx = t; idx < N * N; idx += 1024) {
    const int i = idx >> 9, k = idx & (N - 1);
    s += Yg[idx] * Ygt[(size_t)k * N + i];
  }
  red[t] = s;
  __syncthreads();
  for (int off = 512; off > 0; off >>= 1) {
    if (t < off) red[t] += red[t + off];
    __syncthreads();
  }
  if (t == 0) *s_out = red[0];
}

// -------------------------------------------------- final scalar assembly
// sc: [0]=||G||,[1]=trG,[2]=||Gt||,[3]=trGt,[4]=mse,[5]=cross
__global__ void final_loss_kernel(const float* __restrict__ sc, float* __restrict__ out) {
  const float tr_cross = sqrtf(sc[0]) * sqrtf(sc[2]) * sc[5];
  out[0] = (sc[4] + sc[1] + sc[3] - 2.f * tr_cross) / 512.f;
}

extern "C" void kernel_launch(void* const* d_in, const int* in_sizes, int n_in,
                              void* d_out, int out_size, void* d_ws, size_t ws_size,
                              hipStream_t stream) {
  const float* X  = (const float*)d_in[0];
  const float* Xt = (const float*)d_in[1];
  float* ws = (float*)d_ws;
  const size_t S = (size_t)N * N;

  // ws layout (floats). Every slot is fully written each call before being read.
  float* statsG  = ws + 0;     // [0]=normG [1]=trG
  float* statsGt = ws + 2;     // [0]=normGt [1]=trGt
  float* s_mse   = ws + 4;
  float* s_cross = ws + 5;
  float* psG  = ws + 16;
  float* ptG  = ws + 80;
  float* psGt = ws + 144;
  float* ptGt = ws + 208;
  float* M    = ws + 272;
  float* Mt   = ws + 784;
  float* G    = ws + 2048;
  float* Gt   = G + S;
  float* B0 = Gt + S;
  float* B1 = B0 + S;
  float* B2 = B1 + S;
  float* B3 = B2 + S;
  float* B4 = B3 + S;
  float* B5 = B4 + S;
  float* Spart = B5 + S;       // 2*KS*S floats of split-K partials (~8 MB)
                               // total ws use ~16.8 MB

  row_mean_kernel<<<dim3(N, 2), 256, 0, stream>>>(X, Xt, M, Mt);

  gram_partial_kernel<<<dim3(8, 8, 2 * KS), 128, 0, stream>>>(X, Xt, Spart);
  gram_fixup_kernel<<<dim3(64, 2), 256, 0, stream>>>(Spart, M, Mt, G, Gt,
                                                     psG, ptG, psGt, ptGt);
  finalize_stats_kernel<<<1, 64, 0, stream>>>(psG,  ptG,  statsG);
  finalize_stats_kernel<<<1, 64, 0, stream>>>(psGt, ptGt, statsGt);
  mse_kernel<<<1, 512, 0, stream>>>(M, Mt, s_mse);

  dim3 mgrid(16, 16);
  // ---- Newton-Schulz sqrtm(G)
  float *Y = B0, *Z = B1, *T = B2, *Yn = B3, *Zn = B4;
  init_yz_kernel<<<1024, 256, 0, stream>>>(G, statsG, Y, Z);
  for (int it = 0; it < NS_ITERS; ++it) {
    gemm512_kernel<<<mgrid, 32, 0, stream>>>(Z, Y, T, 0);   // T = 1.5I - 0.5 Z@Y
    gemm512_kernel<<<mgrid, 32, 0, stream>>>(Y, T, Yn, 1);  // Y' = Y@T
    gemm512_kernel<<<mgrid, 32, 0, stream>>>(T, Z, Zn, 1);  // Z' = T@Z
    float* tmp = Y; Y = Yn; Yn = tmp;
    tmp = Z; Z = Zn; Zn = tmp;
  }
  float* Yg = Y;  // = B3 after 15 swaps

  // ---- Newton-Schulz sqrtm(Gt), avoiding Yg's buffer
  float *Y2 = B0, *Z2 = B1, *T2 = B2, *Yn2 = B4, *Zn2 = B5;
  init_yz_kernel<<<1024, 256, 0, stream>>>(Gt, statsGt, Y2, Z2);
  for (int it = 0; it < NS_ITERS; ++it) {
    gemm512_kernel<<<mgrid, 32, 0, stream>>>(Z2, Y2, T2, 0);
    gemm512_kernel<<<mgrid, 32, 0, stream>>>(Y2, T2, Yn2, 1);
    gemm512_kernel<<<mgrid, 32, 0, stream>>>(T2, Z2, Zn2, 1);
    float* tmp = Y2; Y2 = Yn2; Yn2 = tmp;
    tmp = Z2; Z2 = Zn2; Zn2 = tmp;
  }
  float* Ygt = Y2;  // = B4 after 15 swaps

  cross_trace_kernel<<<1, 1024, 0, stream>>>(Yg, Ygt, s_cross);
  final_loss_kernel<<<1, 1, 0, stream>>>(ws, (float*)d_out);
}